// search_54975581389440
// MI455X (gfx1250) — compile-verified
//
#include <hip/hip_runtime.h>
#include <math.h>

// ---------------------------------------------------------------------------
// NCC patch search for MI455X (gfx1250, wave32, WMMA + Tensor Data Mover).
//   A: [16, 64, 96, 96] f32 template
//   B: [16, 64, 159, 159] f32 search domain
//   out[b,i,j] = <A[b], B[b, :, i:i+96, j:j+96]> / sqrt(||A[b]||^2 * ||Bpatch||^2)
//
// Numerator via Toeplitz-form FP32 WMMA (v_wmma_f32_16x16x4_f32):
//   D[i,j] = sum_u AT[i,u] * B[u, j+x],  AT[i,u] = A[u-i, x] (zero-padded).
// B slab staged into LDS with TENSOR_LOAD_TO_LDS (TDM, tracked by TENSORcnt).
// ---------------------------------------------------------------------------

typedef float v2f __attribute__((ext_vector_type(2)));
typedef float v8f __attribute__((ext_vector_type(8)));
typedef unsigned int v4u __attribute__((ext_vector_type(4)));
typedef int v4i __attribute__((ext_vector_type(4)));
typedef int v8i __attribute__((ext_vector_type(8)));

#define NB   16      // batches
#define NC   64      // channels
#define HWA  96      // template h=w
#define HBB  159     // big h=w
#define NS   64      // output h=w (159-96+1)

#define A_STRIDE 129           // LDS column stride for A (mod 64 == 1 -> conflict free)
#define B_STRIDE 112           // LDS row stride for B slab (TDM tile_dim0)
#define A_LDS_FLOATS (96 * A_STRIDE)    // 12384
#define B_LDS_FLOATS (160 * B_STRIDE)   // 17920
#define LDS_BYTES ((A_LDS_FLOATS + B_LDS_FLOATS) * 4)   // 121,216 B < 320 KB/WGP

// Issue a 2-D f32 tile DMA: global (rows x cols, row pitch HBB) -> LDS (pitch 112).
// D# layout per CDNA5 ISA. Wave-uniform; call from one wave.
__device__ __forceinline__ void tdm_load_b_slab(const float* gsrc, unsigned lds_byte_off,
                                                unsigned tensor_cols)
{
    unsigned long long ga = (unsigned long long)(const void*)gsrc;

    v4u g0;
    g0.x = 1u;                                   // count=1, is_restore=0, gather=0
    g0.y = lds_byte_off;                         // lds_addr (bytes)
    g0.z = (unsigned)(ga & 0xFFFFFFFFu);         // global_addr[31:0]
    g0.w = (unsigned)((ga >> 32) & 0x01FFFFFFu)  // global_addr[56:32]
         | (2u << 30);                           // type = 2 ("image")

    v8i g1;
    g1[0] = (int)(2u << 16);                     // workgroup_mask=0, data_size=2 (4B)
    g1[1] = (int)(tensor_cols << 16);            // tensor_dim0[15:0] in bits 63:48
    g1[2] = (int)((unsigned)HBB << 16);          // tensor_dim1[15:0] in bits 95:80
    g1[3] = (int)((unsigned)B_STRIDE << 16);     // tile_dim0 in bits 127:112
    g1[4] = (int)(unsigned)HBB;                  // tile_dim1 = 159 rows (tile_dim2=0)
    g1[5] = (int)(unsigned)HBB;                  // tensor_dim0_stride[31:0] = 159
    g1[6] = (int)((unsigned)(HBB * HBB & 0xFFFF) << 16); // tensor_dim1_stride[15:0]
    g1[7] = (int)((unsigned)(HBB * HBB) >> 16);  // tensor_dim1_stride[47:16]

    v4i gz = {0, 0, 0, 0};                       // groups 2/3 unused (<=2D tensor)
#if defined(__clang_major__) && __clang_major__ >= 23
    v8i gz8 = {0, 0, 0, 0, 0, 0, 0, 0};
    __builtin_amdgcn_tensor_load_to_lds(g0, g1, gz, gz, gz8, 0);
#else
    __builtin_amdgcn_tensor_load_to_lds(g0, g1, gz, gz, 0);
#endif
}

// ---------------------------------------------------------------------------
// Main WMMA kernel: one workgroup per (j-tile, batch, channel-group).
// 128 threads = 4 wave32 waves. Each wave computes out[b, 0:64, j0:j0+16]
// partial-summed over 16 channels (its x-subset), accumulated in VGPRs.
// ---------------------------------------------------------------------------
__global__ __launch_bounds__(128) void ncc_wmma_kernel(
    const float* __restrict__ A, const float* __restrict__ B,
    float* __restrict__ ws_num)
{
    extern __shared__ float smem[];
    float* Ald = smem;                    // [x][ypad] : Ald[x*129 + (y+16)], pads zeroed
    float* Bld = smem + A_LDS_FLOATS;     // [u][v]    : Bld[u*112 + v], row 159 zeroed

    const int jt  = blockIdx.x;           // 0..3  j tile
    const int b   = blockIdx.y;           // 0..15 batch
    const int cg  = blockIdx.z;           // 0..3  channel group (16 channels each)
    const int j0  = jt * 16;
    const int tid = threadIdx.x;
    const int wave = tid >> 5;
    const int lane = tid & 31;
    const int M   = lane & 15;            // row (A) / col (B,C,D) within fragment
    const int K2  = (lane >> 4) << 1;     // K base: lanes 0-15 -> 0, lanes 16-31 -> 2

    // LDS byte offset of the B slab (dynamic LDS begins after static LDS).
    const unsigned bld_off = __builtin_amdgcn_groupstaticsize()
                           + (unsigned)(A_LDS_FLOATS * 4);

    // Zero A zero-pad region (whole plane once; data region overwritten per channel)
    for (int idx = tid; idx < A_LDS_FLOATS; idx += 128) Ald[idx] = 0.0f;
    // Zero B pad row 159 (only ever multiplied by zero A entries; must be finite)
    for (int idx = tid; idx < B_STRIDE; idx += 128) Bld[159 * B_STRIDE + idx] = 0.0f;

    v8f acc0 = {}, acc1 = {}, acc2 = {}, acc3 = {};   // i-tiles 0..3 (i = 16t + m)

    for (int ci = 0; ci < 16; ++ci) {
        const int c = cg * 16 + ci;
        __syncthreads();   // previous compute done before overwriting LDS

        // ---- stage B slab via Tensor Data Mover (wave 0 issues the DMA) ----
        // Tile = 112 cols x 159 rows from tile start (row 0, col j0); cols past
        // the tensor edge (tensor_dim0 = 159 - j0) are zero-filled by TDM OOB.
        if (wave == 0) {
            const float* Bg = B + (size_t)(b * NC + c) * (HBB * HBB) + j0;
            tdm_load_b_slab(Bg, bld_off, (unsigned)(HBB - j0));
        }

        // ---- stage A plane manually (transposed + zero-padded in y),
        //      overlapping with the in-flight TDM transfer ----
        const float* Ag = A + (size_t)(b * NC + c) * (HWA * HWA);
        for (int idx = tid; idx < HWA * HWA; idx += 128) {
            int y = idx / HWA, x = idx - y * HWA;
            Ald[x * A_STRIDE + 16 + y] = Ag[idx];
        }

        if (wave == 0) __builtin_amdgcn_s_wait_tensorcnt(0);  // TDM data in LDS
        __syncthreads();                                      // publish to all waves

        // ---- compute: waves split x; 112 WMMAs per (c,x) per wave ----
        for (int xi = 0; xi < 24; ++xi) {
            const int x = wave + 4 * xi;

            // Build 28 Toeplitz A-fragments: af[s][M,K] = A[c, 4s+K-M, x] (0 outside)
            const float* Acol = Ald + x * A_STRIDE + (16 + K2 - M);
            v2f af[28];
            #pragma unroll
            for (int s = 0; s < 28; ++s) {
                af[s].x = Acol[4 * s];        // K = K2
                af[s].y = Acol[4 * s + 1];    // K = K2 + 1
            }

            // q = u/4 indexes distinct B-fragments; each serves up to 4 i-tiles
            const float* Bcol = Bld + (x + M) + K2 * B_STRIDE;
            #pragma unroll
            for (int q = 0; q < 40; ++q) {
                v2f bf;
                bf.x = Bcol[(4 * q) * B_STRIDE];             // row 4q+K2,   col j0+x+N
                bf.y = Bcol[(4 * q) * B_STRIDE + B_STRIDE];  // row 4q+K2+1
                if (q < 28)
                    acc0 = __builtin_amdgcn_wmma_f32_16x16x4_f32(
                        false, af[q],      false, bf, (short)0, acc0, false, false);
                if (q >= 4 && q < 32)
                    acc1 = __builtin_amdgcn_wmma_f32_16x16x4_f32(
                        false, af[q - 4],  false, bf, (short)0, acc1, false, false);
                if (q >= 8 && q < 36)
                    acc2 = __builtin_amdgcn_wmma_f32_16x16x4_f32(
                        false, af[q - 8],  false, bf, (short)0, acc2, false, false);
                if (q >= 12)
                    acc3 = __builtin_amdgcn_wmma_f32_16x16x4_f32(
                        false, af[q - 12], false, bf, (short)0, acc3, false, false);
            }
        }
    }

    // ---- cross-wave reduction (deterministic, fixed order), reuse B LDS ----
    __syncthreads();
    float* red = Bld;   // needs 4 waves * 1024 floats = 16 KB, fits easily
    {
        v8f accs[4] = {acc0, acc1, acc2, acc3};
        #pragma unroll
        for (int t = 0; t < 4; ++t)
            #pragma unroll
            for (int r = 0; r < 8; ++r)
                red[((wave * 4 + t) * 8 + r) * 32 + lane] = accs[t][r];
    }
    __syncthreads();
    if (wave == 0) {
        #pragma unroll
        for (int t = 0; t < 4; ++t) {
            #pragma unroll
            for (int r = 0; r < 8; ++r) {
                float v = red[((0 * 4 + t) * 8 + r) * 32 + lane]
                        + red[((1 * 4 + t) * 8 + r) * 32 + lane]
                        + red[((2 * 4 + t) * 8 + r) * 32 + lane]
                        + red[((3 * 4 + t) * 8 + r) * 32 + lane];
                int i = 16 * t + r + ((lane >> 4) << 3);   // C/D layout: M = r + 8*(lane/16)
                int j = j0 + (lane & 15);
                ws_num[(((size_t)cg * NB + b) * NS + i) * NS + j] = v;
            }
        }
    }
}

// ---------------------------------------------------------------------------
// Helper kernels (denominator + final reduce): ~1% of total FLOPs.
// ---------------------------------------------------------------------------
__global__ void sumA_kernel(const float* __restrict__ A, float* __restrict__ sumA)
{
    __shared__ float red[256];
    const int b = blockIdx.x;
    const float* p = A + (size_t)b * (NC * HWA * HWA);
    float s = 0.0f;
    for (int idx = threadIdx.x; idx < NC * HWA * HWA; idx += 256) {
        float v = p[idx]; s += v * v;
    }
    red[threadIdx.x] = s; __syncthreads();
    for (int o = 128; o > 0; o >>= 1) {
        if (threadIdx.x < o) red[threadIdx.x] += red[threadIdx.x + o];
        __syncthreads();
    }
    if (threadIdx.x == 0) sumA[b] = red[0];
}

__global__ void sq_kernel(const float* __restrict__ B, float* __restrict__ sq)
{
    int idx = blockIdx.x * blockDim.x + threadIdx.x;
    if (idx >= NB * HBB * HBB) return;
    int b = idx / (HBB * HBB);
    int r = idx - b * (HBB * HBB);
    const float* p = B + (size_t)b * NC * HBB * HBB + r;
    float s = 0.0f;
    for (int c = 0; c < NC; ++c) { float v = p[(size_t)c * HBB * HBB]; s += v * v; }
    sq[idx] = s;
}

// rw[b][u][j] = sum_{x<96} sq[b][u][j+x]
__global__ void rowsum_kernel(const float* __restrict__ sq, float* __restrict__ rw)
{
    int idx = blockIdx.x * blockDim.x + threadIdx.x;
    if (idx >= NB * HBB * NS) return;
    int j = idx & 63;
    int t = idx >> 6;              // = b*159 + u
    int u = t % HBB, b = t / HBB;
    const float* p = sq + ((size_t)b * HBB + u) * HBB + j;
    float s = 0.0f;
    for (int x = 0; x < HWA; ++x) s += p[x];
    rw[idx] = s;
}

__global__ void normalize_kernel(const float* __restrict__ ws_num,
                                 const float* __restrict__ rw,
                                 const float* __restrict__ sumA,
                                 float* __restrict__ out)
{
    int idx = blockIdx.x * blockDim.x + threadIdx.x;
    if (idx >= NB * NS * NS) return;
    int j = idx & 63;
    int i = (idx >> 6) & 63;
    int b = idx >> 12;
    float num = 0.0f;
    for (int cg = 0; cg < 4; ++cg)
        num += ws_num[(((size_t)cg * NB + b) * NS + i) * NS + j];
    float sb = 0.0f;
    const float* p = rw + ((size_t)b * HBB + i) * NS + j;
    for (int y = 0; y < HWA; ++y) sb += p[(size_t)y * NS];
    out[idx] = num / sqrtf(sumA[b] * sb);
}

// ---------------------------------------------------------------------------
extern "C" void kernel_launch(void* const* d_in, const int* in_sizes, int n_in,
                              void* d_out, int out_size, void* d_ws, size_t ws_size,
                              hipStream_t stream)
{
    const float* A = (const float*)d_in[0];   // [16,64,96,96]
    const float* B = (const float*)d_in[1];   // [16,64,159,159]
    float* out = (float*)d_out;               // [16,64,64]
    float* ws = (float*)d_ws;

    // Workspace layout (floats): 16 + 404496 + 162816 + 262144  (~3.2 MB total)
    float* ws_sumA = ws;
    float* ws_sq   = ws_sumA + 16;
    float* ws_rw   = ws_sq + NB * HBB * HBB;       // 404496
    float* ws_num  = ws_rw + NB * HBB * NS;        // 162816

    sumA_kernel<<<NB, 256, 0, stream>>>(A, ws_sumA);

    { int total = NB * HBB * HBB;
      sq_kernel<<<(total + 255) / 256, 256, 0, stream>>>(B, ws_sq); }

    { int total = NB * HBB * NS;
      rowsum_kernel<<<(total + 255) / 256, 256, 0, stream>>>(ws_sq, ws_rw); }

    { dim3 grid(4, NB, 4);   // (j-tile, batch, channel-group) = 256 workgroups
      ncc_wmma_kernel<<<grid, 128, LDS_BYTES, stream>>>(A, B, ws_num); }

    { int total = NB * NS * NS;
      normalize_kernel<<<(total + 255) / 256, 256, 0, stream>>>(ws_num, ws_rw, ws_sumA, out); }
}